// earth_mover_distance_19061064860093
// MI455X (gfx1250) — compile-verified
//
#include <hip/hip_runtime.h>
#include <math.h>

typedef float v2f __attribute__((ext_vector_type(2)));
typedef float v8f __attribute__((ext_vector_type(8)));

#define EMD_EPS 1e-9f
#define NMAX 2048

// select a[h] for h in [0,8) without dynamic indexing (avoids scratch)
__device__ __forceinline__ float sel8(const float a[8], int h) {
    float v = a[0];
#pragma unroll
    for (int k = 1; k < 8; ++k) v = (h == k) ? a[k] : v;
    return v;
}

// One WMMA produces a full 16x16 d2 tile:
// A(16x4) = [-2x1, -2y1, -2z1, 1], B(4x16) = [x2; y2; z2; |p2|^2], C = |p1|^2 per row
// D[i][j] = |p1_i|^2 + |p2_j|^2 - 2 p1_i . p2_j
__device__ __forceinline__ v8f d2_tile(v2f Af, v2f Bf, v8f C) {
    return __builtin_amdgcn_wmma_f32_16x16x4_f32(false, Af, false, Bf,
                                                 (short)0, C, false, false);
}

__global__ __launch_bounds__(256) void emd_init(
    const float* __restrict__ xyz1, const float* __restrict__ xyz2,
    float* __restrict__ remainL, float* __restrict__ remainR,
    float* __restrict__ n1, float* __restrict__ n2,
    float* __restrict__ colW, float* __restrict__ out,
    int b, int n, int m, float multiL, float multiR)
{
    int idx = blockIdx.x * 256 + threadIdx.x;
    if (idx < b * n) {
        const float* p = xyz1 + (size_t)idx * 3;
        float x = p[0], y = p[1], z = p[2];
        n1[idx] = x * x + y * y + z * z;
        remainL[idx] = multiL;
    }
    if (idx < b * m) {
        const float* p = xyz2 + (size_t)idx * 3;
        float x = p[0], y = p[1], z = p[2];
        n2[idx] = x * x + y * y + z * z;
        remainR[idx] = multiR;
        colW[idx] = 0.0f;
    }
    if (idx < b) out[idx] = 0.0f;
}

// Pass A: sumK[j] = sum_i exp(level*d2[i,j]) * remainL[i]  ->  ratioR[j]
// Each wave owns 16 columns (fixed B fragment), loops over all row tiles.
__global__ __launch_bounds__(256) void emd_passA(
    const float* __restrict__ xyz1, const float* __restrict__ xyz2,
    const float* __restrict__ n1g, const float* __restrict__ n2g,
    const float* __restrict__ remainL, const float* __restrict__ remainR,
    float* __restrict__ ratioR, int n, int m, float level)
{
    __shared__ float s_xyz1[NMAX * 3];
    __shared__ float s_n1[NMAX];
    __shared__ float s_rL[NMAX];
    const int bb = blockIdx.y, tid = threadIdx.x;
    const float* x1 = xyz1 + (size_t)bb * n * 3;
    for (int t = tid; t < n * 3; t += 256) s_xyz1[t] = x1[t];
    for (int t = tid; t < n; t += 256) {
        s_n1[t] = n1g[(size_t)bb * n + t];
        s_rL[t] = remainL[(size_t)bb * n + t];
    }
    __syncthreads();

    const int wave = tid >> 5, lane = tid & 31;
    const int h = lane & 15;
    const bool hi = lane >= 16;
    const int j = blockIdx.x * 128 + wave * 16 + h;

    const float* p2 = xyz2 + ((size_t)bb * m + j) * 3;
    v2f Bf;
    Bf.x = hi ? p2[2] : p2[0];
    Bf.y = hi ? n2g[(size_t)bb * m + j] : p2[1];

    float colAcc = 0.0f;
    for (int i0 = 0; i0 < n; i0 += 16) {
        const int row = i0 + h;
        v2f Af;
        Af.x = -2.0f * (hi ? s_xyz1[row * 3 + 2] : s_xyz1[row * 3 + 0]);
        Af.y = hi ? 1.0f : (-2.0f * s_xyz1[row * 3 + 1]);
        const int rbase = i0 + (hi ? 8 : 0);
        v8f C;
#pragma unroll
        for (int v = 0; v < 8; ++v) C[v] = s_n1[rbase + v];
        v8f D = d2_tile(Af, Bf, C);
#pragma unroll
        for (int v = 0; v < 8; ++v) {
            float dv = fmaxf(D[v], 0.0f);
            float K = __expf(level * dv);
            colAcc += K * s_rL[rbase + v];
        }
    }
    colAcc += __shfl_xor(colAcc, 16, 32);   // pair row-halves of same column
    if (!hi) {
        size_t gj = (size_t)bb * m + j;
        ratioR[gj] = remainR[gj] / (EMD_EPS + colAcc);
    }
}

// Pass B: sumKR[i] = sum_j exp(level*d2[i,j]) * ratioR[j]  ->  ratioL[i]
// Each wave owns 16 rows (fixed A fragment + C), loops over all column tiles.
__global__ __launch_bounds__(256) void emd_passB(
    const float* __restrict__ xyz1, const float* __restrict__ xyz2,
    const float* __restrict__ n1g, const float* __restrict__ n2g,
    const float* __restrict__ remainL, const float* __restrict__ ratioR,
    float* __restrict__ ratioL, int n, int m, float level)
{
    __shared__ float s_xyz2[NMAX * 3];
    __shared__ float s_n2[NMAX];
    __shared__ float s_rR[NMAX];
    const int bb = blockIdx.y, tid = threadIdx.x;
    const float* x2 = xyz2 + (size_t)bb * m * 3;
    for (int t = tid; t < m * 3; t += 256) s_xyz2[t] = x2[t];
    for (int t = tid; t < m; t += 256) {
        s_n2[t] = n2g[(size_t)bb * m + t];
        s_rR[t] = ratioR[(size_t)bb * m + t];
    }
    __syncthreads();

    const int wave = tid >> 5, lane = tid & 31;
    const int h = lane & 15;
    const bool hi = lane >= 16;
    const int i0 = blockIdx.x * 128 + wave * 16;
    const int arow = i0 + h;
    const int rbase = i0 + (hi ? 8 : 0);

    const float* p1 = xyz1 + ((size_t)bb * n + arow) * 3;
    v2f Af;
    Af.x = -2.0f * (hi ? p1[2] : p1[0]);
    Af.y = hi ? 1.0f : (-2.0f * p1[1]);
    v8f C0;
#pragma unroll
    for (int v = 0; v < 8; ++v) C0[v] = n1g[(size_t)bb * n + rbase + v];

    float rowAcc[8] = {0, 0, 0, 0, 0, 0, 0, 0};
    for (int j0 = 0; j0 < m; j0 += 16) {
        const int col = j0 + h;
        v2f Bf;
        Bf.x = hi ? s_xyz2[col * 3 + 2] : s_xyz2[col * 3 + 0];
        Bf.y = hi ? s_n2[col] : s_xyz2[col * 3 + 1];
        v8f D = d2_tile(Af, Bf, C0);
        float rr = s_rR[col];
#pragma unroll
        for (int v = 0; v < 8; ++v) {
            float dv = fmaxf(D[v], 0.0f);
            rowAcc[v] += __expf(level * dv) * rr;
        }
    }
#pragma unroll
    for (int v = 0; v < 8; ++v) {      // reduce within each 16-lane half
        float x = rowAcc[v];
        x += __shfl_xor(x, 1, 32);
        x += __shfl_xor(x, 2, 32);
        x += __shfl_xor(x, 4, 32);
        x += __shfl_xor(x, 8, 32);
        rowAcc[v] = x;
    }
    if (h < 8) {
        size_t gi = (size_t)bb * n + rbase + h;
        ratioL[gi] = remainL[gi] / (EMD_EPS + sel8(rowAcc, h));
    }
}

// Pass C: w = K*ratioL*ratioR. Accumulate cost += w*sqrt(d2);
// remainL -= rowsum(w) (clamped); column sums of w -> colW via LDS + global atomics.
__global__ __launch_bounds__(256) void emd_passC(
    const float* __restrict__ xyz1, const float* __restrict__ xyz2,
    const float* __restrict__ n1g, const float* __restrict__ n2g,
    const float* __restrict__ ratioL, const float* __restrict__ ratioR,
    float* __restrict__ remainL, float* __restrict__ colW,
    float* __restrict__ out, int n, int m, float level)
{
    __shared__ float s_xyz2[NMAX * 3];
    __shared__ float s_n2[NMAX];
    __shared__ float s_rR[NMAX];
    __shared__ float s_cw[NMAX];
    const int bb = blockIdx.y, tid = threadIdx.x;
    const float* x2 = xyz2 + (size_t)bb * m * 3;
    for (int t = tid; t < m * 3; t += 256) s_xyz2[t] = x2[t];
    for (int t = tid; t < m; t += 256) {
        s_n2[t] = n2g[(size_t)bb * m + t];
        s_rR[t] = ratioR[(size_t)bb * m + t];
        s_cw[t] = 0.0f;
    }
    __syncthreads();

    const int wave = tid >> 5, lane = tid & 31;
    const int h = lane & 15;
    const bool hi = lane >= 16;
    const int i0 = blockIdx.x * 128 + wave * 16;
    const int arow = i0 + h;
    const int rbase = i0 + (hi ? 8 : 0);

    const float* p1 = xyz1 + ((size_t)bb * n + arow) * 3;
    v2f Af;
    Af.x = -2.0f * (hi ? p1[2] : p1[0]);
    Af.y = hi ? 1.0f : (-2.0f * p1[1]);
    v8f C0;
    float rl[8];
#pragma unroll
    for (int v = 0; v < 8; ++v) {
        C0[v] = n1g[(size_t)bb * n + rbase + v];
        rl[v] = ratioL[(size_t)bb * n + rbase + v];
    }

    float rowAcc[8] = {0, 0, 0, 0, 0, 0, 0, 0};
    float costAcc = 0.0f;
    for (int j0 = 0; j0 < m; j0 += 16) {
        const int col = j0 + h;
        v2f Bf;
        Bf.x = hi ? s_xyz2[col * 3 + 2] : s_xyz2[col * 3 + 0];
        Bf.y = hi ? s_n2[col] : s_xyz2[col * 3 + 1];
        v8f D = d2_tile(Af, Bf, C0);
        float rr = s_rR[col];
        float colPart = 0.0f;
#pragma unroll
        for (int v = 0; v < 8; ++v) {
            float dv = fmaxf(D[v], 0.0f);
            float w = __expf(level * dv) * rl[v] * rr;
            costAcc += w * sqrtf(dv);
            rowAcc[v] += w;
            colPart += w;
        }
        colPart += __shfl_xor(colPart, 16, 32);
        if (!hi) atomicAdd(&s_cw[col], colPart);
    }
#pragma unroll
    for (int v = 0; v < 8; ++v) {
        float x = rowAcc[v];
        x += __shfl_xor(x, 1, 32);
        x += __shfl_xor(x, 2, 32);
        x += __shfl_xor(x, 4, 32);
        x += __shfl_xor(x, 8, 32);
        rowAcc[v] = x;
    }
    if (h < 8) {
        size_t gi = (size_t)bb * n + rbase + h;
        remainL[gi] = fmaxf(0.0f, remainL[gi] - sel8(rowAcc, h));
    }
    costAcc += __shfl_xor(costAcc, 1, 32);
    costAcc += __shfl_xor(costAcc, 2, 32);
    costAcc += __shfl_xor(costAcc, 4, 32);
    costAcc += __shfl_xor(costAcc, 8, 32);
    costAcc += __shfl_xor(costAcc, 16, 32);
    if (lane == 0) atomicAdd(&out[bb], costAcc);
    __syncthreads();
    for (int t = tid; t < m; t += 256)
        atomicAdd(&colW[(size_t)bb * m + t], s_cw[t]);
}

// Pass D: remainR update (clamped) + reset colW for next iteration.
__global__ __launch_bounds__(256) void emd_passD(
    float* __restrict__ remainR, float* __restrict__ colW, int bm)
{
    int idx = blockIdx.x * 256 + threadIdx.x;
    if (idx < bm) {
        remainR[idx] = fmaxf(0.0f, remainR[idx] - colW[idx]);
        colW[idx] = 0.0f;
    }
}

extern "C" void kernel_launch(void* const* d_in, const int* in_sizes, int n_in,
                              void* d_out, int out_size, void* d_ws, size_t ws_size,
                              hipStream_t stream)
{
    const float* xyz1 = (const float*)d_in[0];
    const float* xyz2 = (const float*)d_in[1];
    float* out = (float*)d_out;
    const int b = out_size;
    const int n = in_sizes[0] / (3 * b);
    const int m = in_sizes[1] / (3 * b);

    float* f = (float*)d_ws;
    float* remainL = f; f += (size_t)b * n;
    float* ratioL  = f; f += (size_t)b * n;
    float* n1      = f; f += (size_t)b * n;
    float* remainR = f; f += (size_t)b * m;
    float* ratioR  = f; f += (size_t)b * m;
    float* n2      = f; f += (size_t)b * m;
    float* colW    = f; f += (size_t)b * m;

    const int mL = (m / n) > 1 ? (m / n) : 1;
    const int mR = (n / m) > 1 ? (n / m) : 1;
    const int mx = n > m ? n : m;

    emd_init<<<(b * mx + 255) / 256, 256, 0, stream>>>(
        xyz1, xyz2, remainL, remainR, n1, n2, colW, out,
        b, n, m, (float)mL, (float)mR);

    float levels[10];
    int k = 0;
    for (int j = 7; j >= -1; --j) levels[k++] = -powf(4.0f, (float)j);
    levels[k++] = 0.0f;

    dim3 gA(m / 128, b), gB(n / 128, b);
    for (int it = 0; it < 10; ++it) {
        const float lvl = levels[it];
        emd_passA<<<gA, 256, 0, stream>>>(xyz1, xyz2, n1, n2, remainL, remainR,
                                          ratioR, n, m, lvl);
        emd_passB<<<gB, 256, 0, stream>>>(xyz1, xyz2, n1, n2, remainL, ratioR,
                                          ratioL, n, m, lvl);
        emd_passC<<<gB, 256, 0, stream>>>(xyz1, xyz2, n1, n2, ratioL, ratioR,
                                          remainL, colW, out, n, m, lvl);
        emd_passD<<<((b * m) + 255) / 256, 256, 0, stream>>>(remainR, colW, b * m);
    }
}